// DigitCaps_26044681683574
// MI455X (gfx1250) — compile-verified
//
#include <hip/hip_runtime.h>
#include <hip/hip_bf16.h>

typedef __attribute__((ext_vector_type(2))) float v2f;
typedef __attribute__((ext_vector_type(8))) float v8f;

#define CAPS_B    512
#define CAPS_R    1152
#define CAPS_C    10
#define CAPS_O    16
#define CAPS_I    8
#define MTILE     16
#define WAVES_PER_WG 8
#define RSPLIT    8                                  // r-chunks (workgroups) per batch tile
#define N_BC      (CAPS_B * CAPS_C)                  // 5120 (b,c) pairs
#define SOUT      (CAPS_B * CAPS_C * CAPS_O)         // 81920 floats

// ---------------------------------------------------------------------------
// Shared per-wave routing body.
//
// u_hat tiles are computed TRANSPOSED via V_WMMA_F32_16X16X4_F32:
//   D[16o x 16m] = A(W^T)[16o x 4i] * B(u)[4i x 16m]   (two K=4 chunks, i=0..7)
// In the C/D layout each lane owns column m = lane%16, rows o = {j, j+8*half}:
//   - logit l[m,c] = sum_o u_hatT * vsum  -> per-lane v8f mul + hadd + 1 shfl_xor(16)
//   - softmax over 10 classes is fully per-lane
//   - s accumulation: s_acc[c] += coup[c] * dT[c]  (packed FMA, no cross-lane movs)
// ---------------------------------------------------------------------------
__device__ __forceinline__ void routing_wave_body(
    const float* __restrict__ u_base,   // per-lane: u + (b0+m)*R*8 + 2h
    const float* __restrict__ w_base,   // per-lane: W + 2h*16 + m
    int r_begin, int r_end, bool use_logits,
    const v8f* __restrict__ vsr,        // [CAPS_C], per-lane vsum[m][c][j+8h]
    v8f* __restrict__ s_acc)            // [CAPS_C]
{
    for (int r = r_begin; r < r_end; ++r) {
        // B operand (u), i = {0..3} and {4..7}: element i = kb + vgpr + 2h
        v2f bu_lo = *(const v2f*)(u_base + (size_t)r * CAPS_I + 0);
        v2f bu_hi = *(const v2f*)(u_base + (size_t)r * CAPS_I + 4);

        v8f dT[CAPS_C];
        #pragma unroll
        for (int c = 0; c < CAPS_C; ++c) {
            const float* wp = w_base + (size_t)(r * CAPS_C + c) * (CAPS_I * CAPS_O);
            v2f a_lo, a_hi;
            a_lo.x = wp[0 * CAPS_O];            // i = 0 + 2h
            a_lo.y = wp[1 * CAPS_O];            // i = 1 + 2h
            a_hi.x = wp[4 * CAPS_O];            // i = 4 + 2h
            a_hi.y = wp[5 * CAPS_O];            // i = 5 + 2h
            v8f d = {};
            d = __builtin_amdgcn_wmma_f32_16x16x4_f32(
                    false, a_lo, false, bu_lo, (short)0, d, false, false);
            d = __builtin_amdgcn_wmma_f32_16x16x4_f32(
                    false, a_hi, false, bu_hi, (short)0, d, false, false);
            dT[c] = d;
        }

        float coup[CAPS_C];
        if (use_logits) {
            float l[CAPS_C];
            #pragma unroll
            for (int c = 0; c < CAPS_C; ++c) {
                v8f p = dT[c] * vsr[c];                       // packed muls
                float lp = ((p[0] + p[1]) + (p[2] + p[3])) +
                           ((p[4] + p[5]) + (p[6] + p[7]));
                l[c] = lp + __shfl_xor(lp, 16, 32);           // combine o-halves
            }
            float mx = l[0];
            #pragma unroll
            for (int c = 1; c < CAPS_C; ++c) mx = fmaxf(mx, l[c]);
            float ssum = 0.0f;
            #pragma unroll
            for (int c = 0; c < CAPS_C; ++c) { coup[c] = __expf(l[c] - mx); ssum += coup[c]; }
            float inv = 1.0f / ssum;
            #pragma unroll
            for (int c = 0; c < CAPS_C; ++c) coup[c] *= inv;
        } else {
            #pragma unroll
            for (int c = 0; c < CAPS_C; ++c) coup[c] = 0.1f;  // softmax(0), C=10
        }

        #pragma unroll
        for (int c = 0; c < CAPS_C; ++c)
            s_acc[c] += dT[c] * coup[c];                      // packed FMAs
    }
}

// ---------------------------------------------------------------------------
// Split pipeline (primary path): per routing round, 256 workgroups compute
// partial s sums (LDS ds_add_f32 pre-reduction, then global f32 atomics into
// s_buf), then a tiny squash kernel produces v / accumulates vsum.
// ---------------------------------------------------------------------------
__global__ void zero_ws_kernel(float* __restrict__ p, int n) {
    for (int i = blockIdx.x * blockDim.x + threadIdx.x; i < n;
         i += gridDim.x * blockDim.x)
        p[i] = 0.0f;
}

__global__ __launch_bounds__(256) void routing_partial_kernel(
    const float* __restrict__ u,        // [512,1152,8]
    const float* __restrict__ W,        // [1152,10,8,16]
    const float* __restrict__ vsum_buf, // [512,10,16]
    float* __restrict__ s_buf,          // [512,10,16] (pre-zeroed)
    int iter)
{
    __shared__ __align__(32) float s_red[MTILE][CAPS_C][CAPS_O];  // 10 KB

    const int tid  = threadIdx.x;
    const int lane = tid & 31;
    const int wave = tid >> 5;
    const int m    = lane & 15;
    const int h    = lane >> 4;
    const int b0   = blockIdx.x * MTILE;
    const int rch  = blockIdx.y;                     // 0..RSPLIT-1

    for (int idx = tid; idx < MTILE * CAPS_C * CAPS_O; idx += 256)
        ((float*)s_red)[idx] = 0.0f;
    __syncthreads();

    const float* u_base = u + (size_t)(b0 + m) * (CAPS_R * CAPS_I) + 2 * h;
    const float* w_base = W + (size_t)(2 * h) * CAPS_O + m;

    v8f vsr[CAPS_C];
    if (iter > 0) {
        const float* vp = vsum_buf + (size_t)(b0 + m) * (CAPS_C * CAPS_O) + 8 * h;
        #pragma unroll
        for (int c = 0; c < CAPS_C; ++c)
            vsr[c] = *(const v8f*)(vp + c * CAPS_O);          // 8 contiguous floats
    } else {
        #pragma unroll
        for (int c = 0; c < CAPS_C; ++c) { v8f z = {}; vsr[c] = z; }
    }

    v8f s_acc[CAPS_C];
    #pragma unroll
    for (int c = 0; c < CAPS_C; ++c) { v8f z = {}; s_acc[c] = z; }

    const int rpw = CAPS_R / (WAVES_PER_WG * RSPLIT);         // 18
    const int r_begin = (rch * WAVES_PER_WG + wave) * rpw;
    routing_wave_body(u_base, w_base, r_begin, r_begin + rpw, iter > 0, vsr, s_acc);

    // Cross-wave pre-reduction in LDS (ds_add_f32).
    #pragma unroll
    for (int c = 0; c < CAPS_C; ++c)
        #pragma unroll
        for (int j = 0; j < 8; ++j)
            atomicAdd(&s_red[m][c][j + 8 * h], s_acc[c][j]);
    __syncthreads();

    // Push the workgroup tile to global (RSPLIT-way contention per address).
    for (int idx = tid; idx < MTILE * CAPS_C * CAPS_O; idx += 256) {
        const int m2  = idx / (CAPS_C * CAPS_O);
        const int rem = idx % (CAPS_C * CAPS_O);
        unsafeAtomicAdd(&s_buf[(size_t)(b0 + m2) * (CAPS_C * CAPS_O) + rem],
                        ((float*)s_red)[idx]);
    }
}

__global__ void routing_squash_kernel(
    float* __restrict__ s_buf,     // [512,10,16]; reset to 0 after reading
    float* __restrict__ vsum_buf,  // [512,10,16]; += v
    float* __restrict__ out,       // [512,10,16]; written on final round
    int final_round)
{
    const int t = blockIdx.x * blockDim.x + threadIdx.x;      // (b,c) pair
    if (t >= N_BC) return;
    float* sp = s_buf + (size_t)t * CAPS_O;
    float s[CAPS_O];
    float ss = 0.0f;
    #pragma unroll
    for (int o = 0; o < CAPS_O; ++o) { s[o] = sp[o]; ss = fmaf(s[o], s[o], ss); }
    const float factor = ss / (sqrtf(ss) * (1.0f + ss));
    float* vp = final_round ? (out + (size_t)t * CAPS_O)
                            : (vsum_buf + (size_t)t * CAPS_O);
    #pragma unroll
    for (int o = 0; o < CAPS_O; ++o) {
        const float val = factor * s[o];
        if (final_round) vp[o] = val;
        else             vp[o] += val;    // logit history: b_t = dot(u_hat, sum v)
        sp[o] = 0.0f;                     // reset s for next round
    }
}

// ---------------------------------------------------------------------------
// Fallback: fully fused single kernel (used if d_ws is too small).
// ---------------------------------------------------------------------------
__global__ __launch_bounds__(256) void capsule_routing_fused(
    const float* __restrict__ u, const float* __restrict__ W,
    float* __restrict__ out)
{
    __shared__ __align__(32) float s_red[MTILE][CAPS_C][CAPS_O];
    __shared__ __align__(32) float v_lds[MTILE][CAPS_C][CAPS_O];
    __shared__ __align__(32) float vsum_lds[MTILE][CAPS_C][CAPS_O];

    const int tid  = threadIdx.x;
    const int lane = tid & 31;
    const int wave = tid >> 5;
    const int m    = lane & 15;
    const int h    = lane >> 4;
    const int b0   = blockIdx.x * MTILE;

    for (int idx = tid; idx < MTILE * CAPS_C * CAPS_O; idx += 256) {
        ((float*)s_red)[idx]    = 0.0f;
        ((float*)vsum_lds)[idx] = 0.0f;
    }
    __syncthreads();

    const float* u_base = u + (size_t)(b0 + m) * (CAPS_R * CAPS_I) + 2 * h;
    const float* w_base = W + (size_t)(2 * h) * CAPS_O + m;
    const int rpw = CAPS_R / WAVES_PER_WG;                    // 144
    const int r_begin = wave * rpw;

    for (int iter = 0; iter < 3; ++iter) {
        v8f vsr[CAPS_C];
        if (iter > 0) {
            #pragma unroll
            for (int c = 0; c < CAPS_C; ++c)
                vsr[c] = *(const v8f*)(&vsum_lds[m][c][8 * h]);
        } else {
            #pragma unroll
            for (int c = 0; c < CAPS_C; ++c) { v8f z = {}; vsr[c] = z; }
        }
        v8f s_acc[CAPS_C];
        #pragma unroll
        for (int c = 0; c < CAPS_C; ++c) { v8f z = {}; s_acc[c] = z; }

        routing_wave_body(u_base, w_base, r_begin, r_begin + rpw, iter > 0, vsr, s_acc);

        #pragma unroll
        for (int c = 0; c < CAPS_C; ++c)
            #pragma unroll
            for (int j = 0; j < 8; ++j)
                atomicAdd(&s_red[m][c][j + 8 * h], s_acc[c][j]);
        __syncthreads();

        if (tid < MTILE * CAPS_C) {
            const int m2 = tid / CAPS_C;
            const int c2 = tid % CAPS_C;
            float ss = 0.0f;
            #pragma unroll
            for (int o = 0; o < CAPS_O; ++o) {
                float x = s_red[m2][c2][o];
                ss = fmaf(x, x, ss);
            }
            const float factor = ss / (sqrtf(ss) * (1.0f + ss));
            #pragma unroll
            for (int o = 0; o < CAPS_O; ++o) {
                const float val = factor * s_red[m2][c2][o];
                v_lds[m2][c2][o] = val;
                vsum_lds[m2][c2][o] += val;
            }
        }
        __syncthreads();
        for (int idx = tid; idx < MTILE * CAPS_C * CAPS_O; idx += 256)
            ((float*)s_red)[idx] = 0.0f;
        __syncthreads();
    }

    if (tid < MTILE * CAPS_C) {
        const int m2 = tid / CAPS_C;
        const int c2 = tid % CAPS_C;
        float* op = out + (size_t)(b0 + m2) * (CAPS_C * CAPS_O) + c2 * CAPS_O;
        #pragma unroll
        for (int o = 0; o < CAPS_O; ++o) op[o] = v_lds[m2][c2][o];
    }
}

// ---------------------------------------------------------------------------
extern "C" void kernel_launch(void* const* d_in, const int* in_sizes, int n_in,
                              void* d_out, int out_size, void* d_ws, size_t ws_size,
                              hipStream_t stream) {
    (void)in_sizes; (void)n_in; (void)out_size;
    const float* u = (const float*)d_in[0];   // [512,1152,8]
    const float* W = (const float*)d_in[1];   // [1,1152,10,8,16]
    float* out     = (float*)d_out;           // [512,10,16]

    const size_t ws_needed = (size_t)(2 * SOUT) * sizeof(float);  // 640 KB
    if (d_ws != nullptr && ws_size >= ws_needed) {
        float* s_buf    = (float*)d_ws;
        float* vsum_buf = s_buf + SOUT;
        zero_ws_kernel<<<160, 256, 0, stream>>>(s_buf, 2 * SOUT);
        for (int it = 0; it < 3; ++it) {
            routing_partial_kernel<<<dim3(CAPS_B / MTILE, RSPLIT), 256, 0, stream>>>(
                u, W, vsum_buf, s_buf, it);
            routing_squash_kernel<<<dim3(N_BC / 256), 256, 0, stream>>>(
                s_buf, vsum_buf, out, it == 2 ? 1 : 0);
        }
    } else {
        capsule_routing_fused<<<dim3(CAPS_B / MTILE), 256, 0, stream>>>(u, W, out);
    }
}